// ModelNew_50388556316962
// MI455X (gfx1250) — compile-verified
//
#include <hip/hip_runtime.h>

// D = A(16x4) x B(4x16) + C(16x16), f32 everywhere.
typedef __attribute__((ext_vector_type(2))) float v2f;
typedef __attribute__((ext_vector_type(4))) float v4f;
typedef __attribute__((ext_vector_type(8))) float v8f;

#define N_COLS   8192
#define N_ROWS   1024
#define N_GROUPS 32
#define SCALE    0.75f

#define ROW_TILES   (N_ROWS / 16)   // 64 tiles of 16 rows (WMMA M dim)
#define KSPLIT_BLK  8               // blockIdx.y
#define WAVES_PER_B 8               // 256 threads
#define K_PER_WAVE  (N_COLS / (KSPLIT_BLK * WAVES_PER_B))  // 128
#define K_ITERS     (K_PER_WAVE / 4)                        // 32 wmma steps

// ---------------------------------------------------------------------------
// Kernel 1: w_total[k] = SCALE * sum_g wsums[g][k]; also zero the output
// (harness poisons d_out, and the GEMV kernel accumulates with atomics).
// ---------------------------------------------------------------------------
__global__ void __launch_bounds__(256)
prep_kernel(const float* __restrict__ wsums, float* __restrict__ wtot,
            float* __restrict__ out) {
  int i = blockIdx.x * blockDim.x + threadIdx.x;
  if (i < N_COLS) {
    float s = 0.f;
#pragma unroll
    for (int g = 0; g < N_GROUPS; ++g) s += wsums[g * N_COLS + i];
    wtot[i] = s * SCALE;
  }
  if (i < N_ROWS) out[i] = 0.f;
}

// ---------------------------------------------------------------------------
// Kernel 2: out[16*tile + m] += dot(x[row], wtot[kchunk]) via
// V_WMMA_F32_16X16X4_F32.
//
// A layout (32-bit A, 16x4): lanes 0-15 -> {K0,K1}, lanes 16-31 -> {K2,K3},
// row M = lane & 15.  B is wtot[k..k+3] broadcast into all 16 N columns, so
// every column of D carries the same dot product; row m lives in VGPR (m&7)
// on lane 0 (m<8) / lane 16 (m>=8).
// ---------------------------------------------------------------------------
__global__ void __launch_bounds__(256)
gemv_wmma_kernel(const float* __restrict__ x, const float* __restrict__ wtot,
                 float* __restrict__ out) {
  const int lane    = threadIdx.x & 31;
  const int wave    = threadIdx.x >> 5;                 // 0..7
  const int rowBase = blockIdx.x * 16;                  // row tile
  const int kBase   = (blockIdx.y * WAVES_PER_B + wave) * K_PER_WAVE;

  const int halfSel = (lane >> 4) & 1;                  // 0: K0,1  1: K2,3
  const int row     = rowBase + (lane & 15);
  const float* __restrict__ xrow = x + (size_t)row * N_COLS;

  v8f c = {};

#pragma unroll 4
  for (int it = 0; it < K_ITERS; ++it) {
    const int k = kBase + it * 4;

    // A fragment: this lane's two K-values of its row (16B/row per step,
    // sequential over K -> full cacheline reuse across iterations).
    v2f a = *(const v2f*)(xrow + k + 2 * halfSel);

    // B fragment: wtot[k..k+3], uniform address -> broadcast load; select the
    // K-half for this half-wave and replicate across all N columns.
    v4f wv = *(const v4f*)(wtot + k);
    v2f b;
    b.x = halfSel ? wv.z : wv.x;
    b.y = halfSel ? wv.w : wv.y;

    // (neg_a, A, neg_b, B, c_mod, C, reuse_a, reuse_b)
    c = __builtin_amdgcn_wmma_f32_16x16x4_f32(false, a, false, b,
                                              (short)0, c, false, false);
  }

  // Column 0 of D: rows m=0..7 on lane 0 (VGPR m), rows 8..15 on lane 16.
  if (lane == 0) {
#pragma unroll
    for (int v = 0; v < 8; ++v) atomicAdd(&out[rowBase + v], c[v]);
  } else if (lane == 16) {
#pragma unroll
    for (int v = 0; v < 8; ++v) atomicAdd(&out[rowBase + 8 + v], c[v]);
  }
}

// ---------------------------------------------------------------------------
extern "C" void kernel_launch(void* const* d_in, const int* in_sizes, int n_in,
                              void* d_out, int out_size, void* d_ws, size_t ws_size,
                              hipStream_t stream) {
  const float* x     = (const float*)d_in[0];   // [1024, 8192]
  const float* wsums = (const float*)d_in[1];   // [32, 8192]
  float*       out   = (float*)d_out;           // [1024, 1]
  float*       wtot  = (float*)d_ws;            // [8192] scratch

  prep_kernel<<<dim3(N_COLS / 256), dim3(256), 0, stream>>>(wsums, wtot, out);

  gemv_wmma_kernel<<<dim3(ROW_TILES, KSPLIT_BLK), dim3(256), 0, stream>>>(
      x, wtot, out);
}